// CRFSequenceTaggingHead_6365141532752
// MI455X (gfx1250) — compile-verified
//
#include <hip/hip_runtime.h>
#include <hip/hip_bf16.h>

#define SEQ 512
#define BAT 128
#define DIM 1024
#define TAGS 32

typedef __attribute__((ext_vector_type(2))) float v2f;
typedef __attribute__((ext_vector_type(8))) float v8f;

// Broadcast lane k's value to all lanes (k is a compile-time constant after
// unrolling -> lowers to v_readlane_b32; no LDS, no barrier needed: each
// batch recursion lives in exactly one wave32).
__device__ __forceinline__ float bcast(float x, int k) {
  return __int_as_float(__builtin_amdgcn_readlane(__float_as_int(x), k));
}

// ---------------------------------------------------------------------------
// Kernel 1: emissions[s,b,t] = sum_d features[s,b,d]*W[t,d] + bias[t]
// f32 WMMA 16x16x4 (full f32 precision keeps Viterbi argmax exact).
// Memory-bound: 256 MB feature reads / 23.3 TB/s ~ 11 us floor.
// ---------------------------------------------------------------------------
__global__ __launch_bounds__(128) void emis_wmma(const float* __restrict__ F,
                                                 const float* __restrict__ W,
                                                 const float* __restrict__ bias,
                                                 float* __restrict__ emis) {
  const int lane = threadIdx.x & 31;
  const int wave = threadIdx.x >> 5;
  const int r0 = (blockIdx.x * 4 + wave) * 16;
  const int h = lane >> 4;
  const int lo = lane & 15;
  const float* aptr = F + (size_t)(r0 + lo) * DIM + 2 * h;
  const float* b0p = W + (size_t)lo * DIM + 2 * h;
  const float* b1p = W + (size_t)(16 + lo) * DIM + 2 * h;
  v8f acc0 = {};
  v8f acc1 = {};
#pragma unroll 4
  for (int k = 0; k < DIM; k += 4) {
    v2f a = *(const v2f*)(aptr + k);
    v2f b0 = *(const v2f*)(b0p + k);
    v2f b1 = *(const v2f*)(b1p + k);
    acc0 = __builtin_amdgcn_wmma_f32_16x16x4_f32(false, a, false, b0,
                                                 (short)0, acc0, false, false);
    acc1 = __builtin_amdgcn_wmma_f32_16x16x4_f32(false, a, false, b1,
                                                 (short)0, acc1, false, false);
  }
  const float bv0 = bias[lo];
  const float bv1 = bias[16 + lo];
#pragma unroll
  for (int r = 0; r < 8; ++r) {
    const int row = r0 + r + 8 * h;
    emis[(size_t)row * TAGS + lo] = acc0[r] + bv0;
    emis[(size_t)row * TAGS + 16 + lo] = acc1[r] + bv1;
  }
}

// ---------------------------------------------------------------------------
// Kernel 2: forward recursion. One wave32 per batch (block=32). Transition
// column j register-resident in lane j; cross-lane via v_readlane. Next-step
// emission/mask prefetched to hide L2 latency behind the logsumexp body.
// ---------------------------------------------------------------------------
__global__ __launch_bounds__(32) void alpha_fwd(const float* __restrict__ emis,
                                                const int* __restrict__ mask,
                                                const float* __restrict__ trans,
                                                const float* __restrict__ start,
                                                float* __restrict__ alpha) {
  const int j = threadIdx.x;
  const int b = blockIdx.x;
  float tr[TAGS];  // tr[k] = trans[k][j]
#pragma unroll
  for (int k = 0; k < TAGS; ++k) tr[k] = trans[k * TAGS + j];
  float a = emis[(size_t)b * TAGS + j] + start[j];
  alpha[(size_t)b * TAGS + j] = a;
  float e_nx = emis[((size_t)BAT + b) * TAGS + j];
  float m_nx = (float)mask[BAT + b];
  for (int s = 1; s < SEQ; ++s) {
    const float e = e_nx;
    const float m = m_nx;
    const int sn = (s + 1 < SEQ) ? (s + 1) : (SEQ - 1);
    e_nx = emis[((size_t)sn * BAT + b) * TAGS + j];
    m_nx = (float)mask[sn * BAT + b];
    float vv[TAGS];
    float mx = -3.4e38f;
#pragma unroll
    for (int k = 0; k < TAGS; ++k) {
      vv[k] = bcast(a, k) + tr[k];
      mx = fmaxf(mx, vv[k]);
    }
    float sum = 0.f;
#pragma unroll
    for (int k = 0; k < TAGS; ++k) sum += __expf(vv[k] - mx);
    const float nw = mx + __logf(sum) + e;
    a = m * nw + (1.f - m) * a;
    alpha[((size_t)s * BAT + b) * TAGS + j] = a;
  }
}

// ---------------------------------------------------------------------------
// Kernel 3: z[b] = logsumexp_k(alpha[S-1,b,k] + end[k])
// ---------------------------------------------------------------------------
__global__ __launch_bounds__(128) void z_kernel(const float* __restrict__ alpha,
                                                const float* __restrict__ endp,
                                                float* __restrict__ z) {
  const int b = threadIdx.x;
  const float* a = alpha + ((size_t)(SEQ - 1) * BAT + b) * TAGS;
  float mx = -3.4e38f;
  for (int k = 0; k < TAGS; ++k) mx = fmaxf(mx, a[k] + endp[k]);
  float s = 0.f;
  for (int k = 0; k < TAGS; ++k) s += __expf(a[k] + endp[k] - mx);
  z[b] = mx + __logf(s);
}

// ---------------------------------------------------------------------------
// Kernel 4: backward recursion fused with probs = exp(alpha+beta-z).
// Lane j keeps trans row j in registers (beta needs trans[j][k]).
// ---------------------------------------------------------------------------
__global__ __launch_bounds__(32) void beta_probs(
    const float* __restrict__ emis, const int* __restrict__ mask,
    const float* __restrict__ trans, const float* __restrict__ endp,
    const float* __restrict__ alpha, const float* __restrict__ z,
    float* __restrict__ probs) {
  const int j = threadIdx.x;
  const int b = blockIdx.x;
  float trT[TAGS];  // trT[k] = trans[j][k]
#pragma unroll
  for (int k = 0; k < TAGS; ++k) trT[k] = trans[j * TAGS + k];
  const float zb = z[b];
  float bt = endp[j];
  probs[((size_t)(SEQ - 1) * BAT + b) * TAGS + j] =
      __expf(alpha[((size_t)(SEQ - 1) * BAT + b) * TAGS + j] + bt - zb);
  // prefetched operands for step i
  float e_c = emis[((size_t)(SEQ - 1) * BAT + b) * TAGS + j];
  float m_c = (float)mask[(SEQ - 1) * BAT + b];
  float al_c = alpha[((size_t)(SEQ - 2) * BAT + b) * TAGS + j];
  for (int i = SEQ - 1; i >= 1; --i) {
    const float e = e_c;
    const float m = m_c;
    const float alm = al_c;
    const int in = (i - 1 >= 1) ? (i - 1) : 1;
    e_c = emis[((size_t)in * BAT + b) * TAGS + j];
    m_c = (float)mask[in * BAT + b];
    const int ia = (in - 1 >= 0) ? (in - 1) : 0;
    al_c = alpha[((size_t)ia * BAT + b) * TAGS + j];
    const float t = bt + e;
    float vv[TAGS];
    float mx = -3.4e38f;
#pragma unroll
    for (int k = 0; k < TAGS; ++k) {
      vv[k] = bcast(t, k) + trT[k];
      mx = fmaxf(mx, vv[k]);
    }
    float sum = 0.f;
#pragma unroll
    for (int k = 0; k < TAGS; ++k) sum += __expf(vv[k] - mx);
    const float nw = mx + __logf(sum);
    bt = m * nw + (1.f - m) * bt;
    probs[((size_t)(i - 1) * BAT + b) * TAGS + j] = __expf(alm + bt - zb);
  }
}

// ---------------------------------------------------------------------------
// Kernel 5: Viterbi forward. One wave per batch; history uint8; strict '>'
// keeps first argmax (matches jnp.argmax).
// ---------------------------------------------------------------------------
__global__ __launch_bounds__(32) void vit_fwd(const float* __restrict__ emis,
                                              const int* __restrict__ mask,
                                              const float* __restrict__ trans,
                                              const float* __restrict__ start,
                                              const float* __restrict__ endp,
                                              unsigned char* __restrict__ hist,
                                              int* __restrict__ last) {
  const int j = threadIdx.x;
  const int b = blockIdx.x;
  float tr[TAGS];  // tr[k] = trans[k][j]
#pragma unroll
  for (int k = 0; k < TAGS; ++k) tr[k] = trans[k * TAGS + j];
  float score = start[j] + emis[(size_t)b * TAGS + j];
  float e_nx = emis[((size_t)BAT + b) * TAGS + j];
  int m_nx = mask[BAT + b];
  for (int s = 1; s < SEQ; ++s) {
    const float e = e_nx;
    const int m = m_nx;
    const int sn = (s + 1 < SEQ) ? (s + 1) : (SEQ - 1);
    e_nx = emis[((size_t)sn * BAT + b) * TAGS + j];
    m_nx = mask[sn * BAT + b];
    float best = -3.4e38f;
    int bk = 0;
#pragma unroll
    for (int k = 0; k < TAGS; ++k) {
      const float v = bcast(score, k) + tr[k];
      if (v > best) {
        best = v;
        bk = k;
      }
    }
    hist[((size_t)(s - 1) * BAT + b) * TAGS + j] = (unsigned char)bk;
    if (m) score = best + e;
  }
  const float fin = score + endp[j];
  float best = bcast(fin, 0);
  int bj = 0;
#pragma unroll
  for (int k = 1; k < TAGS; ++k) {
    const float v = bcast(fin, k);
    if (v > best) {
      best = v;
      bj = k;
    }
  }
  if (j == 0) last[b] = bj;
}

// ---------------------------------------------------------------------------
// Kernel 6: backtrack (kernel boundary guarantees hist visibility)
// ---------------------------------------------------------------------------
__global__ __launch_bounds__(128) void vit_back(
    const unsigned char* __restrict__ hist, const int* __restrict__ last,
    float* __restrict__ tags_out) {
  const int b = threadIdx.x;
  int tag = last[b];
  tags_out[(size_t)(SEQ - 1) * BAT + b] = (float)tag;
  for (int s = SEQ - 2; s >= 0; --s) {
    tag = hist[((size_t)s * BAT + b) * TAGS + tag];
    tags_out[(size_t)s * BAT + b] = (float)tag;
  }
}

extern "C" void kernel_launch(void* const* d_in, const int* in_sizes, int n_in,
                              void* d_out, int out_size, void* d_ws,
                              size_t ws_size, hipStream_t stream) {
  const float* F = (const float*)d_in[0];      // (S,B,D)
  const int* mask = (const int*)d_in[1];       // (S,B)
  const float* W = (const float*)d_in[2];      // (T,D)
  const float* bias = (const float*)d_in[3];   // (T,)
  const float* trans = (const float*)d_in[4];  // (T,T)
  const float* start = (const float*)d_in[5];  // (T,)
  const float* endp = (const float*)d_in[6];   // (T,)

  char* ws = (char*)d_ws;
  float* emis = (float*)(ws + 0);                         // 8 MB
  float* alpha = (float*)(ws + 8388608);                  // 8 MB
  float* z = (float*)(ws + 16777216);                     // 512 B
  int* last = (int*)(ws + 16777728);                      // 512 B
  unsigned char* hist = (unsigned char*)(ws + 16778240);  // ~2 MB

  float* probs = (float*)d_out;                    // (S,B,T)
  float* tags = probs + (size_t)SEQ * BAT * TAGS;  // (S,B) as float

  emis_wmma<<<(SEQ * BAT) / 64, 128, 0, stream>>>(F, W, bias, emis);
  alpha_fwd<<<BAT, 32, 0, stream>>>(emis, mask, trans, start, alpha);
  z_kernel<<<1, 128, 0, stream>>>(alpha, endp, z);
  beta_probs<<<BAT, 32, 0, stream>>>(emis, mask, trans, endp, alpha, z, probs);
  vit_fwd<<<BAT, 32, 0, stream>>>(emis, mask, trans, start, endp, hist, last);
  vit_back<<<1, 128, 0, stream>>>(hist, last, tags);
}